// HMM_824633720863
// MI455X (gfx1250) — compile-verified
//
#include <hip/hip_runtime.h>
#include <math.h>

// HMM forward (N=16 states, M=64 symbols) on CDNA5 WMMA, latency-optimized.
// Scaled linear-space recurrence with LAZY renormalization: alpha carried
// unnormalized through f32 WMMAs; running per-column product P tracks the
// scale; log-fold + renormalize only every 4 steps. Per-step critical path:
// permlane16_swap -> 4 chained V_WMMA_F32_16X16X4_F32 -> emission multiply.

#define NSTATES 16
#define MOBS    64

typedef float    v2f __attribute__((ext_vector_type(2)));
typedef float    v8f __attribute__((ext_vector_type(8)));
typedef unsigned v2u __attribute__((ext_vector_type(2)));

#if defined(__has_builtin)
#  if __has_builtin(__builtin_amdgcn_permlane16_swap)
#    define HAVE_PLSWAP 1
#  endif
#endif

__device__ __forceinline__ v8f wmma4(v2f a, v2f b, v8f c) {
  // D = A(16x4) * B(4x16) + C(16x16), f32
  return __builtin_amdgcn_wmma_f32_16x16x4_f32(
      false, a, false, b, (short)0, c, false, false);
}

// In-place half-lane recombine of two D-layout registers into two B-operand
// registers: a' = {a.lo16, b.lo16 moved to hi lanes}, b' = {a.hi16 moved to
// lo lanes, b.hi16}. One v_permlane16_swap_b32 when available.
__device__ __forceinline__ void half_swap(float& a, float& b, int hi) {
#ifdef HAVE_PLSWAP
  v2u r = __builtin_amdgcn_permlane16_swap(
      __float_as_uint(a), __float_as_uint(b), false, false);
  a = __uint_as_float(r.x);
  b = __uint_as_float(r.y);
#else
  float sa = __shfl_xor(a, 16, 32);
  float sb = __shfl_xor(b, 16, 32);
  float na = hi ? sb : a;
  float nb = hi ? b  : sa;
  a = na; b = nb;
#endif
}

// st = p(lane) + p(lane ^ 16) in every lane
__device__ __forceinline__ float xadd16(float p) {
#ifdef HAVE_PLSWAP
  v2u r = __builtin_amdgcn_permlane16_swap(
      __float_as_uint(p), __float_as_uint(p), false, false);
  return __uint_as_float(r.x) + __uint_as_float(r.y);
#else
  return p + __shfl_xor(p, 16, 32);
#endif
}

__global__ __launch_bounds__(32) void HMM_fwd_wmma(
    const int*   __restrict__ x,      // (B, Tstride)
    const int*   __restrict__ Tlen,   // (B,)
    const float* __restrict__ trans,  // (N, N)
    const float* __restrict__ emiss,  // (N, M)
    const float* __restrict__ prior,  // (N,)
    float*       __restrict__ out,    // (B,)
    int Tstride)
{
  __shared__ float sA [NSTATES * NSTATES]; // col-softmax of trans, [i][k]
  __shared__ float sE [NSTATES * MOBS];    // row-softmax of emiss
  __shared__ float sPi[NSTATES];

  const int lane = threadIdx.x & 31;
  const int col  = lane & 15;   // batch column within tile
  const int hi   = lane >> 4;   // lane half
  const int b    = blockIdx.x * 16 + col;

  // ---- one-time setup: softmaxes into LDS (threads 0..15) ----
  if (lane < NSTATES) {
    const int k = lane;                       // trans column softmax (axis=0)
    float mx = -3.402823466e38f;
    for (int i = 0; i < NSTATES; ++i) mx = fmaxf(mx, trans[i * NSTATES + k]);
    float s = 0.f;
    for (int i = 0; i < NSTATES; ++i) s += expf(trans[i * NSTATES + k] - mx);
    float inv = 1.f / s;
    for (int i = 0; i < NSTATES; ++i)
      sA[i * NSTATES + k] = expf(trans[i * NSTATES + k] - mx) * inv;

    const int i = lane;                       // emiss row softmax (axis=1)
    float me = -3.402823466e38f;
    for (int m = 0; m < MOBS; ++m) me = fmaxf(me, emiss[i * MOBS + m]);
    float se = 0.f;
    for (int m = 0; m < MOBS; ++m) se += expf(emiss[i * MOBS + m] - me);
    float ie = 1.f / se;
    for (int m = 0; m < MOBS; ++m) sE[i * MOBS + m] = expf(emiss[i * MOBS + m] - me) * ie;
  }
  if (lane == 0) {
    float mp = -3.402823466e38f;
    for (int i = 0; i < NSTATES; ++i) mp = fmaxf(mp, prior[i]);
    float sp = 0.f;
    for (int i = 0; i < NSTATES; ++i) sp += expf(prior[i] - mp);
    float ip = 1.f / sp;
    for (int i = 0; i < NSTATES; ++i) sPi[i] = expf(prior[i] - mp) * ip;
  }
  __syncthreads();

  // ---- loop-invariant A operands: chunk j covers K = 4j..4j+3 ----
  v2f aop[4];
#pragma unroll
  for (int j = 0; j < 4; ++j)
#pragma unroll
    for (int v = 0; v < 2; ++v)
      aop[j][v] = sA[col * NSTATES + (4 * j + v + 2 * hi)];

  const int Tb = Tlen[b];
  int tmax = Tb;                              // uniform max length in tile
  tmax = max(tmax, __shfl_xor(tmax, 1, 32));
  tmax = max(tmax, __shfl_xor(tmax, 2, 32));
  tmax = max(tmax, __shfl_xor(tmax, 4, 32));
  tmax = max(tmax, __shfl_xor(tmax, 8, 32));
  const int tlast = tmax - 1;

  const long xbase = (long)b * Tstride;

  // ---- t = 0: alpha[i,b] = pi[i] * E[i, x0[b]]  (kept UNNORMALIZED) ----
  int xcur = x[xbase];
  float al[8];
  float ps = 0.f;
#pragma unroll
  for (int r = 0; r < 8; ++r) {
    const int i = r + 8 * hi;
    al[r] = sPi[i] * sE[i * MOBS + xcur];
    ps += al[r];
  }
  float P = xadd16(ps);   // running scale product (per batch column)
  float c = 0.f;          // folded log-scale
  float cfin = 0.f, Pfin = 0.f;
  if (Tb == 1) { cfin = c; Pfin = P; }

  xcur = x[xbase + min(1, tlast)];

  // ---- sequential scan over time ----
  for (int t = 1; t < tmax; ++t) {
    const int xnxt = x[xbase + min(t + 1, tlast)];   // branchless prefetch

    // emission gather for this step (issues early, overlaps WMMA chain)
    float ev[8];
#pragma unroll
    for (int r = 0; r < 8; ++r)
      ev[r] = sE[(r + 8 * hi) * MOBS + xcur];

    // Build four K-chunk B operands from D-layout alpha (4 half-swaps).
    float b00 = al[0], b02 = al[2]; half_swap(b00, b02, hi); // j0[0], j2[0]
    float b01 = al[1], b03 = al[3]; half_swap(b01, b03, hi); // j0[1], j2[1]
    float b10 = al[4], b12 = al[6]; half_swap(b10, b12, hi); // j1[0], j3[0]
    float b11 = al[5], b13 = al[7]; half_swap(b11, b13, hi); // j1[1], j3[1]

    v8f acc = (v8f){0.f, 0.f, 0.f, 0.f, 0.f, 0.f, 0.f, 0.f};
    acc = wmma4(aop[0], (v2f){b00, b01}, acc);
    acc = wmma4(aop[1], (v2f){b10, b11}, acc);
    acc = wmma4(aop[2], (v2f){b02, b03}, acc);
    acc = wmma4(aop[3], (v2f){b12, b13}, acc);

    // emission multiply; partial sum stays OFF the alpha critical path
    float p = 0.f;
#pragma unroll
    for (int r = 0; r < 8; ++r) {
      al[r] = acc[r] * ev[r];
      p += al[r];
    }
    const float st = xadd16(p);
    P *= st;

    // branchless capture of the finishing value for this batch column
    const bool fin = (t == Tb - 1);
    cfin = fin ? c : cfin;
    Pfin = fin ? P : Pfin;

    // lazy renormalization every 4 steps (uniform scalar branch)
    if ((t & 3) == 0) {
      c += __logf(P);
      const float ip = 1.f / P;
#pragma unroll
      for (int r = 0; r < 8; ++r) al[r] *= ip;
      P = 1.f;
    }
    xcur = xnxt;
  }

  if (hi == 0) out[b] = __expf(cfin) * Pfin;   // exp(c + log P)
}

extern "C" void kernel_launch(void* const* d_in, const int* in_sizes, int n_in,
                              void* d_out, int out_size, void* d_ws, size_t ws_size,
                              hipStream_t stream) {
  const int*   x     = (const int*)  d_in[0];
  const int*   Tlen  = (const int*)  d_in[1];
  const float* trans = (const float*)d_in[2];
  const float* emiss = (const float*)d_in[3];
  const float* prior = (const float*)d_in[4];
  float*       out   = (float*)d_out;

  const int B       = in_sizes[1];
  const int Tstride = in_sizes[0] / B;
  const int tiles   = B / 16;

  HMM_fwd_wmma<<<tiles, 32, 0, stream>>>(x, Tlen, trans, emiss, prior, out, Tstride);
}